// InteractionFeatureModel_18691697672517
// MI455X (gfx1250) — compile-verified
//
#include <hip/hip_runtime.h>

// Fused InteractionFeatureModel for MI455X (gfx1250, wave32).
//   out[r] = beta0 + sum_n (Z[r,n] + beta[n]) * X[r,n],  Z = X @ triu(Theta,1)
// FP32 WMMA (v_wmma_f32_16x16x4_f32). X slab staged in LDS; Theta from L2 (4MB).
// K loop split into an unmasked bulk region (k < c0 => k < n structurally) and a
// 16-step diagonal region with branchless v_cndmask masking.

typedef float v2f __attribute__((ext_vector_type(2)));
typedef float v4f __attribute__((ext_vector_type(4)));
typedef float v8f __attribute__((ext_vector_type(8)));

#define P_DIM 1024
#define ROWS_PER_BLOCK 64
#define LDS_STRIDE (P_DIM + 4)                 // +4 floats: bank-conflict-free frag reads
#define RED_OFF (ROWS_PER_BLOCK * LDS_STRIDE)  // reduction scratch after X slab
#define SMEM_FLOATS (RED_OFF + 2 * ROWS_PER_BLOCK)

#define WMMA_F32(A, B, C) \
    __builtin_amdgcn_wmma_f32_16x16x4_f32(false, (A), false, (B), (short)0, (C), false, false)

__global__ __launch_bounds__(256) void interaction_wmma_kernel(
    const float* __restrict__ X,
    const float* __restrict__ beta0p,
    const float* __restrict__ beta,
    const float* __restrict__ Theta,
    float* __restrict__ out)
{
    extern __shared__ float smem[];
    const int tid  = threadIdx.x;
    const int lane = tid & 31;
    // wave-uniform -> force into SGPRs so K loops get scalar control flow
    const int wave = __builtin_amdgcn_readfirstlane(tid >> 5);
    const int rg   = wave & 3;     // row group: 16 rows each
    const int cg   = wave >> 2;    // col group: 64 cols of each 128-col chunk
    const int half = lane >> 4;    // fragment half per WMMA layout
    const int ln   = lane & 15;

    const long rowBase = (long)blockIdx.x * ROWS_PER_BLOCK;
    const float* Xblk = X + rowBase * P_DIM;

    // ---- stage 64 x 1024 X slab into LDS (streamed, nontemporal) ----
    for (int i = tid; i < ROWS_PER_BLOCK * (P_DIM / 4); i += 256) {
        const int r  = i >> 8;            // / (P_DIM/4)
        const int c4 = (i & 255) << 2;
        v4f v = __builtin_nontemporal_load((const v4f*)(Xblk + (long)r * P_DIM + c4));
        *(v4f*)(smem + r * LDS_STRIDE + c4) = v;
    }
    __syncthreads();

    float rowSum[8];
#pragma unroll
    for (int v = 0; v < 8; ++v) rowSum[v] = 0.0f;

    // A-frag (16x4 f32): lane ln holds row rg*16+ln; VGPR pair holds K = k0+2*half, +1
    const float* aBase = smem + (rg * 16 + ln) * LDS_STRIDE + 2 * half;

    for (int chunk = 0; chunk < P_DIM / 128; ++chunk) {
        const int c0 = chunk * 128 + cg * 64;   // this wave's first column (SGPR)
        const int n0 = c0 + ln;                 // tile-0 column for this lane
        v8f acc0 = {}, acc1 = {}, acc2 = {}, acc3 = {};

        // B pointer at row kk = 2*half, column n0; rows advance by 4 per K-step
        const float* tptr = Theta + (long)(2 * half) * P_DIM + n0;

        // ---- bulk region: k0 < c0  =>  kk+1 <= c0-1 < n for every column ----
#pragma unroll 2
        for (int k0 = 0; k0 < c0; k0 += 4) {
            v2f a = *(const v2f*)(aBase + k0);          // ds_load_b64, conflict-free
            v2f b0, b1, b2, b3;
            b0.x = tptr[0];   b0.y = tptr[P_DIM];
            b1.x = tptr[16];  b1.y = tptr[P_DIM + 16];
            b2.x = tptr[32];  b2.y = tptr[P_DIM + 32];
            b3.x = tptr[48];  b3.y = tptr[P_DIM + 48];
            acc0 = WMMA_F32(a, b0, acc0);
            acc1 = WMMA_F32(a, b1, acc1);
            acc2 = WMMA_F32(a, b2, acc2);
            acc3 = WMMA_F32(a, b3, acc3);
            tptr += 4 * P_DIM;
        }

        // ---- diagonal region: k0 in [c0, c0+64), branchless k < n masking ----
        int kk = c0 + 2 * half;
#pragma unroll 4
        for (int k0 = c0; k0 < c0 + 64; k0 += 4) {
            v2f a = *(const v2f*)(aBase + k0);
            // unconditional loads, then register selects (v_cndmask, no branches)
            float x0 = tptr[0],  y0 = tptr[P_DIM];
            float x1 = tptr[16], y1 = tptr[P_DIM + 16];
            float x2 = tptr[32], y2 = tptr[P_DIM + 32];
            float x3 = tptr[48], y3 = tptr[P_DIM + 48];
            v2f b0, b1, b2, b3;
            b0.x = (kk < n0     ) ? x0 : 0.0f;  b0.y = (kk + 1 < n0     ) ? y0 : 0.0f;
            b1.x = (kk < n0 + 16) ? x1 : 0.0f;  b1.y = (kk + 1 < n0 + 16) ? y1 : 0.0f;
            b2.x = (kk < n0 + 32) ? x2 : 0.0f;  b2.y = (kk + 1 < n0 + 32) ? y2 : 0.0f;
            b3.x = (kk < n0 + 48) ? x3 : 0.0f;  b3.y = (kk + 1 < n0 + 48) ? y3 : 0.0f;
            acc0 = WMMA_F32(a, b0, acc0);
            acc1 = WMMA_F32(a, b1, acc1);
            acc2 = WMMA_F32(a, b2, acc2);
            acc3 = WMMA_F32(a, b3, acc3);
            tptr += 4 * P_DIM;
            kk += 4;
        }

        // ---- epilogue: rowSum += (Z + beta[n]) * X[r,n]  (fuses X@beta) ----
        v8f accs[4] = {acc0, acc1, acc2, acc3};
#pragma unroll
        for (int t = 0; t < 4; ++t) {
            const int n  = n0 + t * 16;
            const float bn = beta[n];
            const float* xcol = smem + (rg * 16 + 8 * half) * LDS_STRIDE + n;
#pragma unroll
            for (int v = 0; v < 8; ++v)   // C layout: VGPR v -> row v + 8*half
                rowSum[v] += (accs[t][v] + bn) * xcol[v * LDS_STRIDE];
        }
    }

    // ---- reduce 16 columns-per-lane-group across lanes (wave32 halves) ----
#pragma unroll
    for (int v = 0; v < 8; ++v) {
        float s = rowSum[v];
        s += __shfl_xor(s, 1, 32);
        s += __shfl_xor(s, 2, 32);
        s += __shfl_xor(s, 4, 32);
        s += __shfl_xor(s, 8, 32);   // stays within each 16-lane half
        rowSum[v] = s;
    }
    if (ln == 0) {
#pragma unroll
        for (int v = 0; v < 8; ++v)
            smem[RED_OFF + cg * ROWS_PER_BLOCK + rg * 16 + 8 * half + v] = rowSum[v];
    }
    __syncthreads();

    // combine the two col-group waves, add beta0, store
    if (tid < ROWS_PER_BLOCK) {
        float r = smem[RED_OFF + tid] + smem[RED_OFF + ROWS_PER_BLOCK + tid] + beta0p[0];
        __builtin_nontemporal_store(r, out + rowBase + tid);
    }
}

extern "C" void kernel_launch(void* const* d_in, const int* in_sizes, int n_in,
                              void* d_out, int out_size, void* d_ws, size_t ws_size,
                              hipStream_t stream) {
    (void)n_in; (void)d_ws; (void)ws_size; (void)out_size;
    const float* X      = (const float*)d_in[0];
    const float* beta0  = (const float*)d_in[1];
    const float* beta   = (const float*)d_in[2];
    const float* Theta  = (const float*)d_in[3];
    float* out = (float*)d_out;

    const int N = in_sizes[0] / P_DIM;          // 65536
    const size_t smemBytes = (size_t)SMEM_FLOATS * sizeof(float);  // ~257.5 KB of 320 KB WGP LDS

    hipFuncSetAttribute((const void*)interaction_wmma_kernel,
                        hipFuncAttributeMaxDynamicSharedMemorySize, (int)smemBytes);

    dim3 grid(N / ROWS_PER_BLOCK), block(256);
    interaction_wmma_kernel<<<grid, block, smemBytes, stream>>>(X, beta0, beta, Theta, out);
}